// LSTM_67714454389229
// MI455X (gfx1250) — compile-verified
//
#include <hip/hip_runtime.h>

typedef __attribute__((ext_vector_type(2))) float v2f;
typedef __attribute__((ext_vector_type(8))) float v8f;

#define T_STEPS 512
#define HID 18
#define G4 72          // 4*H
#define KCH 5          // ceil(18/4) K-chunks (K padded to 20)
#define NTILES 5       // ceil(72/16) N-tiles (N padded to 80)
#define ROWS 16        // batch rows per wave (WMMA M)
#define WAVES 4
#define BLOCK (WAVES * 32)
#define HSTR 20        // padded h row stride (K-padding slots stay zero)
#define GSTR 76        // padded gate row stride (avoid 64-bank conflicts)

// 1-instruction v_rcp_f32 instead of IEEE divide sequences
__device__ __forceinline__ float sigm_fast(float x) {
  return __builtin_amdgcn_rcpf(1.f + __expf(-x));
}
__device__ __forceinline__ float tanh_fast(float x) {
  return 1.f - 2.f * __builtin_amdgcn_rcpf(__expf(2.f * x) + 1.f);
}

__global__ __launch_bounds__(BLOCK) void lstm_wmma_kernel(
    const float* __restrict__ inp,   // [B, T, H]
    const float* __restrict__ W_ih,  // [4H, H]
    const float* __restrict__ W_hh,  // [4H, H]
    const float* __restrict__ b_ih,  // [4H]
    const float* __restrict__ b_hh,  // [4H]
    const float* __restrict__ W_lin, // [1, H]
    const float* __restrict__ b_lin, // [1]
    float* __restrict__ out,         // [B, T]
    int B)
{
  // B-fragment layout weights: [kc2(10)][nt(5)][lane(32)] as float2
  __shared__ __align__(16) v2f   sWb[2 * KCH * NTILES * 32];
  __shared__ __align__(16) float sGate[WAVES][ROWS * GSTR];
  __shared__ __align__(16) float sH[WAVES][ROWS * HSTR];
  __shared__ __align__(16) float sWlin[HID];

  const int tid  = threadIdx.x;
  const int wv   = tid >> 5;
  const int lane = tid & 31;
  const int rlo  = lane & 15;
  const int hi   = lane >> 4;

  // ---- preload combined [W_ih; W_hh] transposed into B-fragment layout ----
  for (int e = tid; e < 2 * KCH * NTILES * 32; e += BLOCK) {
    int l   = e & 31;
    int nt  = (e >> 5) % NTILES;
    int kc2 = e / (32 * NTILES);
    const float* W = (kc2 < KCH) ? W_ih : W_hh;
    int kc = (kc2 < KCH) ? kc2 : (kc2 - KCH);
    int n  = nt * 16 + (l & 15);
    int k0 = kc * 4 + ((l >> 4) << 1);
    v2f w;
    w.x = (n < G4 && (k0    ) < HID) ? W[n * HID + k0    ] : 0.f;
    w.y = (n < G4 && (k0 + 1) < HID) ? W[n * HID + k0 + 1] : 0.f;
    sWb[e] = w;
  }
  if (tid < HID) sWlin[tid] = W_lin[tid];
  for (int e = lane; e < ROWS * HSTR; e += 32) sH[wv][e] = 0.f;  // h0 = 0 (+K pad)
  __syncthreads();

  // Per-lane bias per N-tile (D-layout col = rlo within tile, all 8 acc rows)
  float biasv[NTILES];
#pragma unroll
  for (int nt = 0; nt < NTILES; ++nt) {
    int n = nt * 16 + rlo;
    biasv[nt] = (n < G4) ? (b_ih[n] + b_hh[n]) : 0.f;
  }
  const float blin = b_lin[0];

  const int rowBase = (blockIdx.x * WAVES + wv) * ROWS;
  const float* xrow = inp + ((size_t)(rowBase + rlo) * T_STEPS) * HID;

  // Branch-free x fragment loader: clamp address, cndmask the pad lanes
  auto load_xfrag = [&](int t, v2f* f) {
    const float* xr = xrow + t * HID;
#pragma unroll
    for (int kc = 0; kc < KCH; ++kc) {
      int k  = kc * 4 + hi * 2;
      int kl = (k + 1 < HID) ? k : 0;          // compile-time for kc<4
      v2f raw = *(const v2f*)(xr + kl);
      f[kc].x = (k + 1 < HID) ? raw.x : 0.f;
      f[kc].y = (k + 1 < HID) ? raw.y : 0.f;
    }
  };

  float c[9];
#pragma unroll
  for (int e = 0; e < 9; ++e) c[e] = 0.f;  // c0 = 0

  v2f xf[KCH], xn[KCH];
  load_xfrag(0, xf);

  for (int t = 0; t < T_STEPS; ++t) {
    // ---- h fragments from LDS (16x4 f32 A layout; pad slots are zero) ----
    v2f hf[KCH];
#pragma unroll
    for (int kc = 0; kc < KCH; ++kc) {
      int k = kc * 4 + hi * 2;
      hf[kc] = *(const v2f*)&sH[wv][rlo * HSTR + k];
    }

    // Prefetch next timestep's x while the recurrence computes
    int tn = (t + 1 < T_STEPS) ? (t + 1) : t;
    load_xfrag(tn, xn);

    // ---- gates = [x,h] @ Wcomb^T + bias via V_WMMA_F32_16X16X4_F32 ----
#pragma unroll
    for (int nt = 0; nt < NTILES; ++nt) {
      v8f acc;
#pragma unroll
      for (int v = 0; v < 8; ++v) acc[v] = biasv[nt];
#pragma unroll
      for (int kc = 0; kc < KCH; ++kc) {
        v2f wb = sWb[(kc * NTILES + nt) * 32 + lane];
        acc = __builtin_amdgcn_wmma_f32_16x16x4_f32(
            false, xf[kc], false, wb, (short)0, acc, false, false);
      }
#pragma unroll
      for (int kc = 0; kc < KCH; ++kc) {
        v2f wb = sWb[((KCH + kc) * NTILES + nt) * 32 + lane];
        acc = __builtin_amdgcn_wmma_f32_16x16x4_f32(
            false, hf[kc], false, wb, (short)0, acc, false, false);
      }
      int col = nt * 16 + rlo;
      if (col < G4) {   // folds to unconditional for nt < 4
#pragma unroll
        for (int v = 0; v < 8; ++v)
          sGate[wv][(v + hi * 8) * GSTR + col] = acc[v];
      }
    }

    // ---- elementwise cell update: 16 rows x 18 = 288 cells, 9 per lane ----
#pragma unroll
    for (int e = 0; e < 9; ++e) {
      int idx = lane * 9 + e;
      int r = idx / HID;
      int j = idx - r * HID;
      float gi = sGate[wv][r * GSTR + j];
      float gf = sGate[wv][r * GSTR + HID + j];
      float gg = sGate[wv][r * GSTR + 2 * HID + j];
      float go = sGate[wv][r * GSTR + 3 * HID + j];
      float cn = sigm_fast(gf) * c[e] + sigm_fast(gi) * tanh_fast(gg);
      c[e] = cn;
      sH[wv][r * HSTR + j] = sigm_fast(go) * tanh_fast(cn);
    }

    // ---- output projection out[b,t] = h . W_lin + b_lin ----
    if (lane < 16) {
      float acc0 = blin;
#pragma unroll
      for (int j = 0; j < HID; ++j)
        acc0 += sH[wv][lane * HSTR + j] * sWlin[j];
      out[(size_t)(rowBase + lane) * T_STEPS + t] = acc0;
    }

    // rotate x double-buffer
#pragma unroll
    for (int kc = 0; kc < KCH; ++kc) xf[kc] = xn[kc];
  }
}

extern "C" void kernel_launch(void* const* d_in, const int* in_sizes, int n_in,
                              void* d_out, int out_size, void* d_ws, size_t ws_size,
                              hipStream_t stream) {
  (void)n_in; (void)out_size; (void)d_ws; (void)ws_size;
  const float* inp   = (const float*)d_in[0];
  const float* W_ih  = (const float*)d_in[1];
  const float* W_hh  = (const float*)d_in[2];
  const float* b_ih  = (const float*)d_in[3];
  const float* b_hh  = (const float*)d_in[4];
  const float* W_lin = (const float*)d_in[5];
  const float* b_lin = (const float*)d_in[6];
  float* out = (float*)d_out;

  const int B = in_sizes[0] / (T_STEPS * HID);  // 4096
  dim3 grid(B / (ROWS * WAVES));                // 64 blocks of 4 waves
  lstm_wmma_kernel<<<grid, BLOCK, 0, stream>>>(
      inp, W_ih, W_hh, b_ih, b_hh, W_lin, b_lin, out, B);
}